// Sinkhorn_Net_19524921328279
// MI455X (gfx1250) — compile-verified
//
#include <hip/hip_runtime.h>

#define N       128
#define LATENT  128
#define BATCH   512
#define NITERS  10
#define LDS_STRIDE 129   // bank(i,j) = (i+j) & 63 -> conflict-free rows AND columns

typedef __attribute__((ext_vector_type(2))) float v2f;
typedef __attribute__((ext_vector_type(8))) float v8f;

// hardware transcendentals: v_exp_f32 (exp2) and v_log_f32 (log2)
#define EXP2F(x) __builtin_amdgcn_exp2f(x)
#define LOG2F(x) __builtin_amdgcn_logf(x)

__global__ __launch_bounds__(128) void sinkhorn_kernel(
    const float* __restrict__ x,      // [B, n]
    const float* __restrict__ w1,     // [latent]
    const float* __restrict__ b1,     // [latent]
    const float* __restrict__ w2,     // [n, latent]
    const float* __restrict__ b2,     // [n]
    const float* __restrict__ noise,  // [S*B, n, n]
    float* __restrict__ out)          // [S*B, n, n] (transposed exp)
{
    extern __shared__ float la[];             // [N][LDS_STRIDE]
    const int m    = blockIdx.x;              // s*B + b
    const int b    = m & (BATCH - 1);
    const int tid  = threadIdx.x;             // 0..127
    const int wave = tid >> 5;                // 0..3
    const int lane = tid & 31;
    const int half = lane >> 4;               // lane group (0: lanes 0-15, 1: lanes 16-31)
    const int l16  = lane & 15;

    const float* np = noise + (size_t)m * (N * N);

    // Prefetch this WG's 64KB noise tile into cache while we do the WMMA GEMM.
    #pragma unroll
    for (int q = 0; q < 4; ++q)
        __builtin_prefetch(np + ((tid << 2) + q) * 32, 0, 1);

    // ---------------- MLP GEMM via V_WMMA_F32_16X16X4_F32 ----------------
    // C[i,j] = sum_l relu(x[b,i]*w1[l]+b1[l]) * w2[j,l]
    // wave owns C rows [32*wave, 32*wave+32): 2 tile-rows x 8 tile-cols.
    const int row0 = wave * 32;
    const float x0 = x[b * N + row0 + l16];
    const float x1 = x[b * N + row0 + 16 + l16];

    v8f acc[2][8];
    #pragma unroll
    for (int r = 0; r < 2; ++r)
        #pragma unroll
        for (int t = 0; t < 8; ++t)
            acc[r][t] = (v8f){0.f, 0.f, 0.f, 0.f, 0.f, 0.f, 0.f, 0.f};

    #pragma unroll 4
    for (int k = 0; k < 32; ++k) {
        const int l0 = 4 * k + 2 * half;   // this lane's K indices: l0, l0+1
        const v2f w1p = *(const v2f*)(w1 + l0);
        const v2f b1p = *(const v2f*)(b1 + l0);
        v2f a0, a1;                        // A frags (16x4 f32 layout)
        a0.x = fmaxf(fmaf(x0, w1p.x, b1p.x), 0.f);
        a0.y = fmaxf(fmaf(x0, w1p.y, b1p.y), 0.f);
        a1.x = fmaxf(fmaf(x1, w1p.x, b1p.x), 0.f);
        a1.y = fmaxf(fmaf(x1, w1p.y, b1p.y), 0.f);
        #pragma unroll
        for (int t = 0; t < 8; ++t) {
            // B frag: B[k_loc, nn] = w2[16t+nn, l0 + v]
            const v2f bf = *(const v2f*)(w2 + (t * 16 + l16) * LATENT + l0);
            acc[0][t] = __builtin_amdgcn_wmma_f32_16x16x4_f32(
                false, a0, false, bf, (short)0, acc[0][t], false, false);
            acc[1][t] = __builtin_amdgcn_wmma_f32_16x16x4_f32(
                false, a1, false, bf, (short)0, acc[1][t], false, false);
        }
    }
    // C/D layout: vgpr e -> M = e + 8*half, N = l16
    #pragma unroll
    for (int r = 0; r < 2; ++r)
        #pragma unroll
        for (int t = 0; t < 8; ++t)
            #pragma unroll
            for (int e = 0; e < 8; ++e) {
                const int row = row0 + 16 * r + e + 8 * half;
                const int col = t * 16 + l16;
                la[row * LDS_STRIDE + col] = acc[r][t][e];
            }
    __syncthreads();

    // ------------- add b2 + gumbel noise, move to log2 space -------------
    const float LOG2E = 1.4426950408889634f;
    const float bb = b2[tid];
    #pragma unroll 4
    for (int i = 0; i < N; ++i) {
        float* p = la + i * LDS_STRIDE + tid;       // bank (i+tid)&63: conflict-free
        *p = (*p + bb + np[i * N + tid]) * LOG2E;   // TEMP=1, NOISE_FACTOR=1
    }
    __syncthreads();

    // ---------------- 10 Sinkhorn iterations, LDS-resident ----------------
    const float NEG_INF = -3.402823466e+38f;
    float vv[N];
    for (int it = 0; it < NITERS; ++it) {
        // row normalize (axis=2): thread owns row 'tid'
        {
            float* rowp = la + tid * LDS_STRIDE;
            float mx = NEG_INF;
            #pragma unroll
            for (int j = 0; j < N; ++j) { vv[j] = rowp[j]; mx = fmaxf(mx, vv[j]); }
            float s = 0.f;
            #pragma unroll
            for (int j = 0; j < N; ++j) s += EXP2F(vv[j] - mx);
            const float lse = mx + LOG2F(s);
            #pragma unroll
            for (int j = 0; j < N; ++j) rowp[j] = vv[j] - lse;
        }
        __syncthreads();
        // column normalize (axis=1): thread owns column 'tid'
        {
            float* colp = la + tid;
            float mx = NEG_INF;
            #pragma unroll
            for (int i = 0; i < N; ++i) { vv[i] = colp[i * LDS_STRIDE]; mx = fmaxf(mx, vv[i]); }
            float s = 0.f;
            #pragma unroll
            for (int i = 0; i < N; ++i) s += EXP2F(vv[i] - mx);
            const float lse = mx + LOG2F(s);
            #pragma unroll
            for (int i = 0; i < N; ++i) colp[i * LDS_STRIDE] = vv[i] - lse;
        }
        __syncthreads();
    }

    // ------- out[m, i, j] = exp(la[j, i])  (transpose + exp, coalesced) -------
    float* op = out + (size_t)m * (N * N);
    #pragma unroll 4
    for (int i = 0; i < N; ++i)
        op[i * N + tid] = EXP2F(la[tid * LDS_STRIDE + i]);
}

extern "C" void kernel_launch(void* const* d_in, const int* in_sizes, int n_in,
                              void* d_out, int out_size, void* d_ws, size_t ws_size,
                              hipStream_t stream) {
    const float* x     = (const float*)d_in[0];
    const float* w1    = (const float*)d_in[1];
    const float* b1    = (const float*)d_in[2];
    const float* w2    = (const float*)d_in[3];
    const float* b2    = (const float*)d_in[4];
    const float* noise = (const float*)d_in[5];
    float* out = (float*)d_out;

    const int n_mats = out_size / (N * N);              // S*B = 2560
    const size_t lds_bytes = N * LDS_STRIDE * sizeof(float);  // 66048 B (< 320KB/WG on CDNA5)
    sinkhorn_kernel<<<n_mats, 128, lds_bytes, stream>>>(x, w1, b1, w2, b2, noise, out);
}